// GNN_13451837571172
// MI455X (gfx1250) — compile-verified
//
#include <hip/hip_runtime.h>

typedef float v2f __attribute__((ext_vector_type(2)));
typedef float v8f __attribute__((ext_vector_type(8)));

// ---------------------------------------------------------------------------
// Degree / normalization kernels
// ---------------------------------------------------------------------------
__global__ void k_init_deg(unsigned* __restrict__ deg, int n) {
    int i = blockIdx.x * blockDim.x + threadIdx.x;
    if (i < n) deg[i] = 0u;
}

__global__ void k_count_deg(const long long* __restrict__ dst,
                            unsigned* __restrict__ deg, int E) {
    int i = blockIdx.x * blockDim.x + threadIdx.x;
    if (i < E) atomicAdd(&deg[(int)dst[i]], 1u);
}

__global__ void k_dinv(const unsigned* __restrict__ deg,
                       float* __restrict__ dinv, int n) {
    int i = blockIdx.x * blockDim.x + threadIdx.x;
    if (i < n) dinv[i] = rsqrtf((float)deg[i] + 1.0f);  // +1 = self loop
}

// ---------------------------------------------------------------------------
// Pack W[K x N] into row-pair interleaved Wt so a B fragment is ONE b64 load:
//   Wt[(k/2)*N + col] = { W[k][col], W[k+1][col] }
// ---------------------------------------------------------------------------
template <int K, int N>
__global__ void k_pack_w(const float* __restrict__ W, float* __restrict__ Wt) {
    int i = blockIdx.x * blockDim.x + threadIdx.x;   // over (K/2)*N
    if (i >= (K / 2) * N) return;
    int k2  = i / N;
    int col = i % N;
    float2 p;
    p.x = W[(size_t)(2 * k2) * N + col];
    p.y = W[(size_t)(2 * k2 + 1) * N + col];
    ((float2*)Wt)[i] = p;
}

// ---------------------------------------------------------------------------
// fp32 WMMA GEMM:  C[M x N] = A[M x K] @ W[K x N]   (W given packed as Wt)
// One block = one 16-row M tile; wave w computes N-tile at n0 = 16*w.
// A tile staged in LDS with padded stride (K+4) -> conflict-free ds_load_2addr.
// Optional fused ReLU on A (used for layer 2 reading layer-1 aggregate).
// ---------------------------------------------------------------------------
template <int K, int N, bool RELU>
__global__ __launch_bounds__(32 * (N / 16))
void gcn_gemm_wmma(const float* __restrict__ A, const float* __restrict__ Wt,
                   float* __restrict__ C, int M) {
    constexpr int LDA = K + 4;              // padded row stride (floats)
    __shared__ float sA[16 * LDA];

    const int tid = threadIdx.x;
    const int m0  = blockIdx.x * 16;

    // Stage 16 x K A-tile (coalesced float4), fuse ReLU if requested.
    {
        const float4* A4 = (const float4*)A;
        for (int i = tid; i < (16 * K) / 4; i += blockDim.x) {
            int row = i / (K / 4);
            int cv  = i % (K / 4);
            int grow = m0 + row;
            if (grow >= M) grow = M - 1;    // clamp (harmless duplicate)
            float4 v = A4[(size_t)grow * (K / 4) + cv];
            if (RELU) {
                v.x = fmaxf(v.x, 0.f); v.y = fmaxf(v.y, 0.f);
                v.z = fmaxf(v.z, 0.f); v.w = fmaxf(v.w, 0.f);
            }
            *(float4*)&sA[row * LDA + cv * 4] = v;
        }
    }
    __syncthreads();

    const int wave = tid >> 5;
    const int lane = tid & 31;
    const int col  = lane & 15;      // N index within tile / M index for A frag
    const int half = lane >> 4;      // selects K pair in A/B fragments
    const int n0   = wave * 16;

    const v2f* Wt2 = (const v2f*)Wt;

    v8f acc = {0.f, 0.f, 0.f, 0.f, 0.f, 0.f, 0.f, 0.f};

#pragma unroll
    for (int k0 = 0; k0 < K; k0 += 4) {
        // A fragment: lane holds A[M=col][kb], A[M=col][kb+1], kb = k0+2*half
        v2f a = *(const v2f*)&sA[col * LDA + k0 + 2 * half];
        // B fragment: one 64-bit load of the packed row pair
        v2f b = Wt2[(size_t)(k0 / 2 + half) * N + n0 + col];
        acc = __builtin_amdgcn_wmma_f32_16x16x4_f32(
            false, a, false, b, (short)0, acc, false, false);
    }

    // C/D layout: VGPR r -> row M = r + 8*half, col = lane&15
#pragma unroll
    for (int r = 0; r < 8; ++r) {
        int grow = m0 + r + 8 * half;
        if (grow < M) C[(size_t)grow * N + n0 + col] = acc[r];
    }
}

// ---------------------------------------------------------------------------
// agg[n][:] = h[n][:] * dinv[n]^2 + bias   (self-loop contribution + bias)
// ---------------------------------------------------------------------------
template <int F>
__global__ void k_init_agg(const float* __restrict__ h,
                           const float* __restrict__ dinv,
                           const float* __restrict__ bias,
                           float* __restrict__ agg, int n) {
    constexpr int FV = F / 4;
    int i = blockIdx.x * blockDim.x + threadIdx.x;
    if (i >= n * FV) return;
    int node = i / FV;
    int cv   = i % FV;
    float d  = dinv[node];
    float d2 = d * d;
    float4 v = ((const float4*)h)[i];
    float4 b = ((const float4*)bias)[cv];
    float4 o;
    o.x = v.x * d2 + b.x; o.y = v.y * d2 + b.y;
    o.z = v.z * d2 + b.z; o.w = v.w * d2 + b.w;
    ((float4*)agg)[i] = o;
}

// ---------------------------------------------------------------------------
// Edge scatter: agg[dst] += h[src] * dinv[src] * dinv[dst]
// F/4 threads per edge, float4 gather, native fp32 atomics.
// ---------------------------------------------------------------------------
template <int F>
__global__ void k_scatter(const long long* __restrict__ src,
                          const long long* __restrict__ dst,
                          const float* __restrict__ h,
                          const float* __restrict__ dinv,
                          float* __restrict__ agg, int E) {
    constexpr int TPE = F / 4;
    long long tid = (long long)blockIdx.x * blockDim.x + threadIdx.x;
    int e = (int)(tid / TPE);
    if (e >= E) return;
    int q = (int)(tid % TPE);
    int s = (int)src[e];
    int d = (int)dst[e];
    float norm = dinv[s] * dinv[d];
    float4 v = ((const float4*)h)[(size_t)s * TPE + q];
    float* base = agg + (size_t)d * F + q * 4;
    unsafeAtomicAdd(base + 0, v.x * norm);
    unsafeAtomicAdd(base + 1, v.y * norm);
    unsafeAtomicAdd(base + 2, v.z * norm);
    unsafeAtomicAdd(base + 3, v.w * norm);
}

// ---------------------------------------------------------------------------
extern "C" void kernel_launch(void* const* d_in, const int* in_sizes, int n_in,
                              void* d_out, int out_size, void* d_ws, size_t ws_size,
                              hipStream_t stream) {
    const float*     x   = (const float*)d_in[0];
    const long long* ei  = (const long long*)d_in[1];
    const float*     W1  = (const float*)d_in[2];
    const float*     b1  = (const float*)d_in[3];
    const float*     W2  = (const float*)d_in[4];
    const float*     b2  = (const float*)d_in[5];
    float*           out = (float*)d_out;

    constexpr int DIN = 128, DHID = 128, DOUT = 64;
    const int N = in_sizes[0] / DIN;     // 50000
    const int E = in_sizes[1] / 2;       // 600000
    const long long* srcp = ei;
    const long long* dstp = ei + E;

    // workspace carve-out (256B aligned)
    char* ws = (char*)d_ws;
    size_t off = 0;
    auto carve = [&](size_t bytes) -> void* {
        void* p = ws + off;
        off = (off + bytes + 255) & ~(size_t)255;
        return p;
    };
    unsigned* deg  = (unsigned*)carve((size_t)N * 4);
    float*    dinv = (float*)   carve((size_t)N * 4);
    float*    h1   = (float*)   carve((size_t)N * DHID * 4);
    float*    agg1 = (float*)   carve((size_t)N * DHID * 4);
    float*    h2   = (float*)   carve((size_t)N * DOUT * 4);
    float*    Wt1  = (float*)   carve((size_t)DIN * DHID * 4);
    float*    Wt2  = (float*)   carve((size_t)DHID * DOUT * 4);

    const int TB = 256;
    auto cdiv = [](long long a, long long b) { return (int)((a + b - 1) / b); };

    // ---- degrees & normalization ----
    k_init_deg <<<cdiv(N, TB), TB, 0, stream>>>(deg, N);
    k_count_deg<<<cdiv(E, TB), TB, 0, stream>>>(dstp, deg, E);
    k_dinv     <<<cdiv(N, TB), TB, 0, stream>>>(deg, dinv, N);

    // ---- pack weights into WMMA B-fragment-friendly row-pair layout ----
    k_pack_w<DIN, DHID><<<cdiv((DIN / 2) * DHID, TB), TB, 0, stream>>>(W1, Wt1);
    k_pack_w<DHID, DOUT><<<cdiv((DHID / 2) * DOUT, TB), TB, 0, stream>>>(W2, Wt2);

    // ---- layer 1: h1 = x @ W1 ----
    gcn_gemm_wmma<DIN, DHID, false>
        <<<cdiv(N, 16), 32 * (DHID / 16), 0, stream>>>(x, Wt1, h1, N);
    // agg1 = selfloop + bias, then edge scatter
    k_init_agg<DHID><<<cdiv((long long)N * (DHID / 4), TB), TB, 0, stream>>>(
        h1, dinv, b1, agg1, N);
    k_scatter<DHID><<<cdiv((long long)E * (DHID / 4), TB), TB, 0, stream>>>(
        srcp, dstp, h1, dinv, agg1, E);

    // ---- layer 2: h2 = relu(agg1) @ W2 (ReLU fused into A staging) ----
    gcn_gemm_wmma<DHID, DOUT, true>
        <<<cdiv(N, 16), 32 * (DOUT / 16), 0, stream>>>(agg1, Wt2, h2, N);
    k_init_agg<DOUT><<<cdiv((long long)N * (DOUT / 4), TB), TB, 0, stream>>>(
        h2, dinv, b2, out, N);
    k_scatter<DOUT><<<cdiv((long long)E * (DOUT / 4), TB), TB, 0, stream>>>(
        srcp, dstp, h2, dinv, out, E);
}